// GAT2_30468497998103
// MI455X (gfx1250) — compile-verified
//
#include <hip/hip_runtime.h>
#include <hip/hip_bf16.h>
#include <cstddef>

typedef __attribute__((ext_vector_type(2))) float v2f;
typedef __attribute__((ext_vector_type(8))) float v8f;

#define F_IN   256
#define H1DIM  128   // 4 heads * 32
#define NHEAD  4
#define HC     32
#define NCLS   40
#define NCLSP  48    // padded to 3 WMMA tiles
#define NEG_SLOPE 0.2f
#define BN_EPS 1e-5f

// ---------- order-preserving float<->uint for atomic max ----------
__device__ __forceinline__ unsigned encf(float f) {
    unsigned u = __float_as_uint(f);
    return (u & 0x80000000u) ? ~u : (u | 0x80000000u);
}
__device__ __forceinline__ float decf(unsigned u) {
    return (u & 0x80000000u) ? __uint_as_float(u ^ 0x80000000u)
                             : __uint_as_float(~u);
}

// src/dst for edge e; e >= E0 are the appended self-loops
__device__ __forceinline__ void edge_sd(const long long* __restrict__ ei,
                                        int e, int E0, int& s, int& d) {
    if (e < E0) { s = (int)ei[e]; d = (int)ei[(size_t)E0 + e]; }
    else        { s = e - E0; d = s; }
}

// ---------------------------------------------------------------- utils
__global__ void k_zero(float* __restrict__ p, int n) {
    int t = blockIdx.x * blockDim.x + threadIdx.x;
    if (t < n) p[t] = 0.0f;
}

// ---------------------------------------------------------------- GEMM1
// h1[N,128] = x[N,256] @ W1[256,128], fp32 WMMA 16x16x4.
// block = 256 threads = 8 waves; wave w -> 16-col tile w; 16-row A panel in LDS.
__global__ void k_gemm1(const float* __restrict__ x, const float* __restrict__ W,
                        float* __restrict__ h) {
    __shared__ float As[16 * 260];              // padded: bank = (4*row + k) & 63
    const int row0 = blockIdx.x << 4;
    {
        const float4* xv = (const float4*)(x + (size_t)row0 * F_IN);
        for (int i = threadIdx.x; i < 16 * 64; i += 256) {
            int r = i >> 6, cc = i & 63;        // 64 float4 per row
            *(float4*)&As[r * 260 + (cc << 2)] = xv[i];
        }
    }
    __syncthreads();
    const int lane = threadIdx.x & 31;
    const int wave = threadIdx.x >> 5;
    const int mrow = lane & 15;
    const int kh   = (lane >> 4) << 1;          // 0 or 2
    const int col  = (wave << 4) + mrow;
    v8f acc = {};
    #pragma unroll 4
    for (int k = 0; k < F_IN; k += 4) {
        v2f a, b;
        a.x = As[mrow * 260 + k + kh];
        a.y = As[mrow * 260 + k + kh + 1];
        b.x = W[(size_t)(k + kh)     * H1DIM + col];
        b.y = W[(size_t)(k + kh + 1) * H1DIM + col];
        acc = __builtin_amdgcn_wmma_f32_16x16x4_f32(false, a, false, b,
                                                    (short)0, acc, false, false);
    }
    const int mbase = row0 + ((lane >> 4) << 3);
    #pragma unroll
    for (int r = 0; r < 8; ++r)
        h[(size_t)(mbase + r) * H1DIM + col] = acc[r];
}

// ---------------------------------------------------------------- GEMM2
// z[N,48] = h2[N,128] @ Wp[128,48]; block = 96 threads = 3 waves.
__global__ void k_gemm2(const float* __restrict__ h2, const float* __restrict__ Wp,
                        float* __restrict__ z) {
    __shared__ float As[16 * 132];
    const int row0 = blockIdx.x << 4;
    {
        const float4* xv = (const float4*)(h2 + (size_t)row0 * H1DIM);
        for (int i = threadIdx.x; i < 16 * 32; i += 96) {
            int r = i >> 5, cc = i & 31;        // 32 float4 per row
            *(float4*)&As[r * 132 + (cc << 2)] = xv[i];
        }
    }
    __syncthreads();
    const int lane = threadIdx.x & 31;
    const int wave = threadIdx.x >> 5;
    const int mrow = lane & 15;
    const int kh   = (lane >> 4) << 1;
    const int col  = (wave << 4) + mrow;
    v8f acc = {};
    #pragma unroll 4
    for (int k = 0; k < H1DIM; k += 4) {
        v2f a, b;
        a.x = As[mrow * 132 + k + kh];
        a.y = As[mrow * 132 + k + kh + 1];
        b.x = Wp[(k + kh)     * NCLSP + col];
        b.y = Wp[(k + kh + 1) * NCLSP + col];
        acc = __builtin_amdgcn_wmma_f32_16x16x4_f32(false, a, false, b,
                                                    (short)0, acc, false, false);
    }
    const int mbase = row0 + ((lane >> 4) << 3);
    #pragma unroll
    for (int r = 0; r < 8; ++r)
        z[(size_t)(mbase + r) * NCLSP + col] = acc[r];
}

// ---------------------------------------------------------------- attention dots
__global__ void k_alpha1(const float* __restrict__ h1, const float* __restrict__ a_src,
                         const float* __restrict__ a_dst, float* __restrict__ as1,
                         float* __restrict__ ad1, int total) {
    int t = blockIdx.x * blockDim.x + threadIdx.x;
    if (t >= total) return;
    int n = t >> 2, hd = t & 3;
    const float* hp = h1 + (size_t)n * H1DIM + hd * HC;
    const float* sp = a_src + hd * HC;
    const float* dp = a_dst + hd * HC;
    float s = 0.f, d = 0.f;
    #pragma unroll
    for (int c = 0; c < HC; ++c) { float v = hp[c]; s += v * sp[c]; d += v * dp[c]; }
    as1[t] = s; ad1[t] = d;
}

__global__ void k_alpha2(const float* __restrict__ z, const float* __restrict__ a_src,
                         const float* __restrict__ a_dst, float* __restrict__ as2,
                         float* __restrict__ ad2, int N) {
    int n = blockIdx.x * blockDim.x + threadIdx.x;
    if (n >= N) return;
    const float* zp = z + (size_t)n * NCLSP;
    float s = 0.f, d = 0.f;
    #pragma unroll
    for (int c = 0; c < NCLS; ++c) { float v = zp[c]; s += v * a_src[c]; d += v * a_dst[c]; }
    as2[n] = s; ad2[n] = d;
}

// ---------------------------------------------------------------- edge passes, layer 1 (H=4)
__global__ void k_edge1_max(const long long* __restrict__ ei, int E0,
                            const float* __restrict__ as1, const float* __restrict__ ad1,
                            float* __restrict__ e1, unsigned* __restrict__ m1u, int total) {
    int t = blockIdx.x * blockDim.x + threadIdx.x;
    if (t >= total) return;
    int e = t >> 2, hd = t & 3, s, d;
    edge_sd(ei, e, E0, s, d);
    float v = as1[s * 4 + hd] + ad1[d * 4 + hd];
    v = v > 0.f ? v : NEG_SLOPE * v;
    e1[t] = v;
    atomicMax(&m1u[d * 4 + hd], encf(v));
}

__global__ void k_edge1_expsum(const long long* __restrict__ ei, int E0,
                               float* __restrict__ e1, const unsigned* __restrict__ m1u,
                               float* __restrict__ den1, int total) {
    int t = blockIdx.x * blockDim.x + threadIdx.x;
    if (t >= total) return;
    int e = t >> 2, hd = t & 3, s, d;
    edge_sd(ei, e, E0, s, d);
    float ex = expf(e1[t] - decf(m1u[d * 4 + hd]));
    e1[t] = ex;                                     // reuse buffer as ex
    unsafeAtomicAdd(&den1[d * 4 + hd], ex);
}

__global__ void k_edge1_agg(const long long* __restrict__ ei, int E0,
                            const float* __restrict__ h1, const float* __restrict__ ex,
                            const float* __restrict__ den, float* __restrict__ out1,
                            int total) {
    int t = blockIdx.x * blockDim.x + threadIdx.x;
    if (t >= total) return;
    int e = t >> 7, c = t & 127, hd = c >> 5, s, d;
    edge_sd(ei, e, E0, s, d);
    float alpha = ex[e * 4 + hd] / den[d * 4 + hd];
    unsafeAtomicAdd(&out1[(size_t)d * H1DIM + c], h1[(size_t)s * H1DIM + c] * alpha);
}

// ---------------------------------------------------------------- batchnorm + ELU
__global__ void k_bn_stats(const float* __restrict__ out1, const float* __restrict__ b1,
                           float* __restrict__ bns, float* __restrict__ bns2,
                           int N, int rowsPerBlk) {
    int c  = threadIdx.x;                           // 128 channels
    int r0 = blockIdx.x * rowsPerBlk;
    int r1 = min(r0 + rowsPerBlk, N);
    float b = b1[c], s = 0.f, s2 = 0.f;
    for (int r = r0; r < r1; ++r) {
        float v = out1[(size_t)r * H1DIM + c] + b;
        s += v; s2 += v * v;
    }
    unsafeAtomicAdd(&bns[c], s);
    unsafeAtomicAdd(&bns2[c], s2);
}

__global__ void k_bn_elu(float* __restrict__ out1, const float* __restrict__ b1,
                         const float* __restrict__ bns, const float* __restrict__ bns2,
                         const float* __restrict__ gamma, const float* __restrict__ beta,
                         int N, int total) {
    int t = blockIdx.x * blockDim.x + threadIdx.x;
    if (t >= total) return;
    int c = t & 127;
    float invN = 1.0f / (float)N;
    float mu  = bns[c] * invN;
    float var = bns2[c] * invN - mu * mu;
    float y = (out1[t] + b1[c] - mu) * rsqrtf(var + BN_EPS) * gamma[c] + beta[c];
    out1[t] = y > 0.f ? y : expm1f(y);
}

// ---------------------------------------------------------------- layer-2 helpers
__global__ void k_padW2(const float* __restrict__ W2, float* __restrict__ Wp, int total) {
    int t = blockIdx.x * blockDim.x + threadIdx.x;
    if (t >= total) return;
    int r = t / NCLSP, c = t - r * NCLSP;
    Wp[t] = (c < NCLS) ? W2[r * NCLS + c] : 0.0f;
}

__global__ void k_edge2_max(const long long* __restrict__ ei, int E0,
                            const float* __restrict__ as2, const float* __restrict__ ad2,
                            float* __restrict__ e2, unsigned* __restrict__ m2u, int total) {
    int e = blockIdx.x * blockDim.x + threadIdx.x;
    if (e >= total) return;
    int s, d; edge_sd(ei, e, E0, s, d);
    float v = as2[s] + ad2[d];
    v = v > 0.f ? v : NEG_SLOPE * v;
    e2[e] = v;
    atomicMax(&m2u[d], encf(v));
}

__global__ void k_edge2_expsum(const long long* __restrict__ ei, int E0,
                               float* __restrict__ e2, const unsigned* __restrict__ m2u,
                               float* __restrict__ den2, int total) {
    int e = blockIdx.x * blockDim.x + threadIdx.x;
    if (e >= total) return;
    int s, d; edge_sd(ei, e, E0, s, d);
    float ex = expf(e2[e] - decf(m2u[d]));
    e2[e] = ex;
    unsafeAtomicAdd(&den2[d], ex);
}

__global__ void k_edge2_agg(const long long* __restrict__ ei, int E0,
                            const float* __restrict__ z, const float* __restrict__ ex,
                            const float* __restrict__ den, float* __restrict__ out,
                            int total) {
    int t = blockIdx.x * blockDim.x + threadIdx.x;
    if (t >= total) return;
    int e = t / NCLS, c = t - e * NCLS, s, d;
    edge_sd(ei, e, E0, s, d);
    float alpha = ex[e] / den[d];
    unsafeAtomicAdd(&out[(size_t)d * NCLS + c], z[(size_t)s * NCLSP + c] * alpha);
}

__global__ void k_bias2(float* __restrict__ out, const float* __restrict__ b2, int total) {
    int t = blockIdx.x * blockDim.x + threadIdx.x;
    if (t >= total) return;
    out[t] += b2[t % NCLS];
}

// ================================================================ launch
extern "C" void kernel_launch(void* const* d_in, const int* in_sizes, int n_in,
                              void* d_out, int out_size, void* d_ws, size_t ws_size,
                              hipStream_t stream) {
    const float*     x      = (const float*)d_in[0];
    const long long* ei     = (const long long*)d_in[1];   // int64 [2,E]
    const float*     W1     = (const float*)d_in[2];
    const float*     a_src1 = (const float*)d_in[3];
    const float*     a_dst1 = (const float*)d_in[4];
    const float*     b1     = (const float*)d_in[5];
    const float*     gamma1 = (const float*)d_in[6];
    const float*     beta1  = (const float*)d_in[7];
    const float*     W2     = (const float*)d_in[8];
    const float*     a_src2 = (const float*)d_in[9];
    const float*     a_dst2 = (const float*)d_in[10];
    const float*     b2     = (const float*)d_in[11];
    float* out = (float*)d_out;

    const int N  = in_sizes[0] / F_IN;   // 50000
    const int E0 = in_sizes[1] / 2;      // 800000
    const int ET = E0 + N;               // + self loops

    // ---- workspace layout (floats) ----
    float* w = (float*)d_ws;
    size_t off = 0;
    float* h1  = w + off; off += (size_t)N * H1DIM;
    float* as1 = w + off; off += (size_t)N * 4;
    float* ad1 = w + off; off += (size_t)N * 4;
    float* e1  = w + off; off += (size_t)ET * 4;
    float* z   = w + off; off += (size_t)N * NCLSP;
    float* as2 = w + off; off += N;
    float* ad2 = w + off; off += N;
    float* e2  = w + off; off += ET;
    float* Wp  = w + off; off += H1DIM * NCLSP;
    // zero-initialized region (contiguous)
    float*    zr   = w + off;
    unsigned* m1u  = (unsigned*)(w + off); off += (size_t)N * 4;  // enc(f) >= 0 for all finite f
    float*    den1 = w + off; off += (size_t)N * 4;
    float*    out1 = w + off; off += (size_t)N * H1DIM;
    float*    bns  = w + off; off += H1DIM;
    float*    bns2 = w + off; off += H1DIM;
    unsigned* m2u  = (unsigned*)(w + off); off += N;
    float*    den2 = w + off; off += N;
    const int zr_n = (int)((w + off) - zr);

    auto cdiv = [](long long a, long long b) { return (int)((a + b - 1) / b); };

    // init
    k_zero<<<cdiv(zr_n, 256), 256, 0, stream>>>(zr, zr_n);
    k_zero<<<cdiv((long long)N * NCLS, 256), 256, 0, stream>>>(out, N * NCLS);

    // ---- layer 1 ----
    k_gemm1<<<N / 16, 256, 0, stream>>>(x, W1, h1);
    k_alpha1<<<cdiv((long long)N * 4, 256), 256, 0, stream>>>(h1, a_src1, a_dst1, as1, ad1, N * 4);
    k_edge1_max   <<<cdiv((long long)ET * 4, 256), 256, 0, stream>>>(ei, E0, as1, ad1, e1, m1u, ET * 4);
    k_edge1_expsum<<<cdiv((long long)ET * 4, 256), 256, 0, stream>>>(ei, E0, e1, m1u, den1, ET * 4);
    k_edge1_agg   <<<cdiv((long long)ET * 128, 256), 256, 0, stream>>>(ei, E0, h1, e1, den1, out1, ET * 128);

    const int rpb = (N + 199) / 200;
    k_bn_stats<<<200, 128, 0, stream>>>(out1, b1, bns, bns2, N, rpb);
    k_bn_elu<<<cdiv((long long)N * H1DIM, 256), 256, 0, stream>>>(out1, b1, bns, bns2,
                                                                  gamma1, beta1, N, N * H1DIM);

    // ---- layer 2 ----
    k_padW2<<<cdiv(H1DIM * NCLSP, 256), 256, 0, stream>>>(W2, Wp, H1DIM * NCLSP);
    k_gemm2<<<N / 16, 96, 0, stream>>>(out1, Wp, z);
    k_alpha2<<<cdiv(N, 256), 256, 0, stream>>>(z, a_src2, a_dst2, as2, ad2, N);
    k_edge2_max   <<<cdiv(ET, 256), 256, 0, stream>>>(ei, E0, as2, ad2, e2, m2u, ET);
    k_edge2_expsum<<<cdiv(ET, 256), 256, 0, stream>>>(ei, E0, e2, m2u, den2, ET);
    k_edge2_agg   <<<cdiv((long long)ET * NCLS, 256), 256, 0, stream>>>(ei, E0, z, e2, den2, out, ET * NCLS);
    k_bias2<<<cdiv((long long)N * NCLS, 256), 256, 0, stream>>>(out, b2, N * NCLS);
}